// LegendreProjection_48644799594943
// MI455X (gfx1250) — compile-verified
//
#include <hip/hip_runtime.h>

typedef __attribute__((ext_vector_type(2))) float v2f;
typedef __attribute__((ext_vector_type(4))) float v4f;
typedef __attribute__((ext_vector_type(8))) float v8f;

#define DEG1 128               // MAX_DEGREE + 1
#define PAIRS_PER_BLOCK 128    // 8 waves * 16 pairs
#define THREADS 256

// ---------------------------------------------------------------------------
// Kernel 1: build W[k][j] = coefficient of T_k in P_j (transposed connection
// matrix, row-major over k).  b_k = C(2k,k)/4^k computed in f64 for accuracy.
// ---------------------------------------------------------------------------
__global__ void legendre_build_connection(float* __restrict__ W) {
    __shared__ float bsh[DEG1];
    if (threadIdx.x == 0) {
        double b = 1.0;
        bsh[0] = 1.0f;
        for (int k = 1; k < DEG1; ++k) {
            b *= (double)(2 * k - 1) / (double)(2 * k);
            bsh[k] = (float)b;
        }
    }
    __syncthreads();
    int j = threadIdx.x;           // launched with 128 threads
    if (j < DEG1) {
        for (int k = 0; k < DEG1; ++k) {
            float c = 0.0f;
            if (k <= j && (((j ^ k) & 1) == 0)) {
                float f = bsh[(j - k) >> 1] * bsh[(j + k) >> 1];
                c = (k == 0) ? f : 2.0f * f;
            }
            W[k * DEG1 + j] = c;
        }
    }
}

// ---------------------------------------------------------------------------
// Kernel 2: per (point,dim) pair, generate Chebyshev T_0..T_127 in-register
// and multiply by the connection matrix with V_WMMA_F32_16X16X4_F32.
// Out[pair, 0..127] = sum_k T[pair,k] * W[k, deg].
//
// LDS holds W row-pair interleaved:  Wp[(k>>1)*128 + j] = (W[k][j], W[k+1][j])
// so each B fragment is ONE contiguous ds_load_b64 into an even VGPR pair
// (B layout: VGPR0 = rows K+0|K+2 by lane half, VGPR1 = rows K+1|K+3).
// nt unrolled x4 -> 4 independent accumulator chains, DS loads prefetched.
// ---------------------------------------------------------------------------
__global__ void __launch_bounds__(THREADS)
legendre_wmma(const float* __restrict__ x, const float* __restrict__ W,
              float* __restrict__ out) {
    __shared__ v2f Wp[DEG1 * DEG1 / 2];        // 64 KB of 320 KB/WGP

    __builtin_prefetch(W, 0, 1);               // global_prefetch_b8 (L2 warm)

    // Cooperative load + row-pair interleave: b128 global loads, b32 ds stores.
    {
        float* Wl = (float*)Wp;
        #pragma unroll
        for (int i = 0; i < (DEG1 * DEG1) / (4 * THREADS); ++i) {
            const int e4 = (threadIdx.x + i * THREADS) * 4;   // j multiple of 4
            const v4f w  = *(const v4f*)(W + e4);
            const int k  = e4 >> 7;
            const int j  = e4 & 127;
            const int b  = ((k >> 1) << 8) + (j << 1) + (k & 1);
            Wl[b]     = w.x;
            Wl[b + 2] = w.y;
            Wl[b + 4] = w.z;
            Wl[b + 6] = w.w;
        }
    }
    __syncthreads();

    const int lane = threadIdx.x & 31;
    const int wave = threadIdx.x >> 5;
    const bool hi  = lane >= 16;               // lanes 16..31 carry K+2,K+3
    const int  hib = hi ? 1 : 0;
    const int  col = lane & 15;

    const long pb = (long)blockIdx.x * PAIRS_PER_BLOCK + (long)wave * 16;

    // pair p = n*2 + d indexes x flat directly ([N,2] row-major)
    const float xs = 2.0f * x[pb + col] - 1.0f;

    // A fragments: afrag[kk] covers K = 4*kk .. 4*kk+3 (16x4 f32 layout:
    // VGPR0 = K0|K2 split by lane half, VGPR1 = K1|K3).
    v2f afrag[32];
    {
        float t0 = 1.0f, t1 = xs;              // T_0, T_1
        #pragma unroll
        for (int kk = 0; kk < 32; ++kk) {
            float t2 = 2.0f * xs * t1 - t0;
            float t3 = 2.0f * xs * t2 - t1;
            v2f a;
            a.x = hi ? t2 : t0;
            a.y = hi ? t3 : t1;
            afrag[kk] = a;
            t0 = 2.0f * xs * t3 - t2;          // T_{4kk+4}
            t1 = 2.0f * xs * t0 - t3;          // T_{4kk+5}
        }
    }

    // C layout: VGPR i -> (pair pb+i, lanes 0-15) / (pair pb+i+8, lanes 16-31),
    // N = degree within tile -> two contiguous 64B runs per b32 store.
    float* outp = out + (pb + (hi ? 8 : 0)) * (long)DEG1 + col;

    #pragma unroll 4
    for (int nt = 0; nt < 8; ++nt) {           // 8 degree tiles of 16
        v8f acc = {};
        #pragma unroll
        for (int kk = 0; kk < 32; ++kk) {      // K = 128 in steps of 4
            const v2f b = Wp[(2 * kk + hib) * 128 + nt * 16 + col];
            acc = __builtin_amdgcn_wmma_f32_16x16x4_f32(
                /*neg_a=*/false, afrag[kk], /*neg_b=*/false, b,
                /*c_mod=*/(short)0, acc, /*reuse_a=*/false, /*reuse_b=*/false);
        }
        #pragma unroll
        for (int i = 0; i < 8; ++i)
            outp[i * DEG1 + nt * 16] = acc[i];
    }
}

// ---------------------------------------------------------------------------
// Tail: plain Legendre recurrence for any pairs not covered by full blocks
// (not used at N = 524288, kept for generality).
// ---------------------------------------------------------------------------
__global__ void legendre_tail(const float* __restrict__ x,
                              float* __restrict__ out,
                              long start, long total) {
    long p = start + (long)blockIdx.x * blockDim.x + threadIdx.x;
    if (p >= total) return;
    float xs = 2.0f * x[p] - 1.0f;
    float pm = 1.0f, pc = xs;
    float* o = out + p * (long)DEG1;
    o[0] = pm;
    o[1] = pc;
    for (int n = 1; n < DEG1 - 1; ++n) {
        float pn = ((2.0f * n + 1.0f) * xs * pc - (float)n * pm) / (float)(n + 1);
        o[n + 1] = pn;
        pm = pc;
        pc = pn;
    }
}

extern "C" void kernel_launch(void* const* d_in, const int* in_sizes, int n_in,
                              void* d_out, int out_size, void* d_ws, size_t ws_size,
                              hipStream_t stream) {
    const float* x   = (const float*)d_in[0];
    float*       out = (float*)d_out;
    float*       W   = (float*)d_ws;           // 64 KB connection matrix

    const long pairs = (long)in_sizes[0];      // N * INPUT_DIM (flat count)

    legendre_build_connection<<<1, 128, 0, stream>>>(W);

    const long full = pairs / PAIRS_PER_BLOCK;
    if (full > 0)
        legendre_wmma<<<(int)full, THREADS, 0, stream>>>(x, W, out);

    const long rem = pairs - full * PAIRS_PER_BLOCK;
    if (rem > 0)
        legendre_tail<<<(int)((rem + 255) / 256), 256, 0, stream>>>(
            x, out, full * PAIRS_PER_BLOCK, pairs);
}